// FreqModModel_25082609009197
// MI455X (gfx1250) — compile-verified
//
#include <hip/hip_runtime.h>
#include <stdint.h>

// ---------------------------------------------------------------------------
// FreqMod synth for MI455X (gfx1250).
//   grid = 128 blocks (one per batch*channel row), block = 1024 threads.
//   - TDM DMAs the 128KB noise row into LDS (overlapped with atan2 frame prep)
//   - three fused prefix sums: thread-serial(32) -> wave shfl scan(32)
//     -> inter-wave scan via chained v_wmma_f32_16x16x4_f32 (triangular L16)
//   - channel mean folded in with global_atomic_add_f32 (out pre-zeroed)
// ---------------------------------------------------------------------------

typedef float    v2f  __attribute__((ext_vector_type(2)));
typedef float    v8f  __attribute__((ext_vector_type(8)));
typedef uint32_t u32x4 __attribute__((ext_vector_type(4)));
typedef int      i32x4 __attribute__((ext_vector_type(4)));
typedef int      i32x8 __attribute__((ext_vector_type(8)));

#define N_SAMPLES 32768
#define N_FRAMES  128
#define THREADS   1024
#define SPT       32          // samples per thread (1024*32 = 32768)

#define MIN_FREQ   0.0036281179138322f     // 40/11025
#define FREQ_RANGE 0.3591836734693878f     // 3960/11025
#define INV_PI     0.3183098861837907f

// Flat pointers to LDS carry the byte offset in addr[31:0] (shared aperture
// occupies the high half) -> truncation yields the wave-relative LDS address
// the TDM descriptor wants.
__device__ __forceinline__ uint32_t lds_byte_offset(const void* p) {
  return (uint32_t)(uintptr_t)p;
}

extern __shared__ float s_noise[];  // N_SAMPLES floats = 128 KB (dynamic LDS)

__global__ __launch_bounds__(THREADS)
void fm_synth_kernel(const float* __restrict__ x,
                     const float* __restrict__ noise,
                     float* __restrict__ out) {
  __shared__ float s_amp[N_FRAMES], s_f0[N_FRAMES], s_mix[N_FRAMES],
                   s_std[N_FRAMES], s_modf[N_FRAMES], s_bb[N_FRAMES];
  __shared__ float s_wtot[3][32];   // per-wave chunk totals, 3 scanned quantities
  __shared__ float s_wscan[3][32];  // half-inclusive scans from the WMMA

  const int tid  = threadIdx.x;
  const int blk  = blockIdx.x;      // 0..127 == b*8 + c
  const int b    = blk >> 3;
  const int lane = tid & 31;
  const int wave = tid >> 5;

  // ---- 1) TDM: DMA this row's 32768 noise floats into LDS (async) ---------
  if (tid < 32) {                   // wave 0 issues; EXEC all ones; uniform args
    uint64_t gaddr = (uint64_t)(uintptr_t)(noise + (size_t)blk * N_SAMPLES);
    uint32_t laddr = lds_byte_offset(s_noise);
    u32x4 g0; i32x8 g1; i32x4 g2 = {0,0,0,0}; i32x4 g3 = {0,0,0,0};
    i32x8 g4 = {0,0,0,0,0,0,0,0};
    // D# group0: count=1 | lds_addr | global_addr[56:0] | type=2
    g0[0] = 1u;
    g0[1] = laddr;
    g0[2] = (uint32_t)gaddr;
    g0[3] = ((uint32_t)(gaddr >> 32) & 0x01FFFFFFu) | (2u << 30);
    // D# group1: data_size=4B, tensor_dim0=tile_dim0=32768 (1-D), stride=32768
    g1[0] = (int)(2u << 16);                  // data_size = 2 (4 bytes)
    g1[1] = (int)(((uint32_t)N_SAMPLES & 0xFFFFu) << 16);  // tensor_dim0 lo
    g1[2] = (int)(1u << 16);                  // tensor_dim0 hi=0 | tensor_dim1=1
    g1[3] = (int)(((uint32_t)N_SAMPLES & 0xFFFFu) << 16);  // tile_dim0
    g1[4] = 0;                                // tile_dim1=0 (unused), tile_dim2=0
    g1[5] = N_SAMPLES;                        // tensor_dim0_stride lo
    g1[6] = 0;
    g1[7] = 0;
    // clang-23 / therock-10.0 lane: 6-arg form
    __builtin_amdgcn_tensor_load_to_lds(g0, g1, g2, g3, g4, 0);
  }

  // ---- 2) frame parameters (overlaps the TDM) -----------------------------
  if (tid < N_FRAMES) {
    const float* xf = x + (size_t)blk * 10 * N_FRAMES + tid;
    float x0 = xf[0*N_FRAMES], x1 = xf[1*N_FRAMES];
    float x2 = xf[2*N_FRAMES], x3 = xf[3*N_FRAMES];
    float x4 = xf[4*N_FRAMES], x5 = xf[5*N_FRAMES];
    float x6 = xf[6*N_FRAMES], x7 = xf[7*N_FRAMES];
    float x8 = xf[8*N_FRAMES], x9 = xf[9*N_FRAMES];
    s_amp[tid]  = sqrtf(x0*x0 + x1*x1);
    s_f0[tid]   = MIN_FREQ + atan2f(x1, x0) * (INV_PI * FREQ_RANGE);
    s_mix[tid]  = 1.0f     + atan2f(x3, x2) * (INV_PI * 0.5f);
    s_std[tid]  = MIN_FREQ + atan2f(x5, x4) * (INV_PI * FREQ_RANGE);
    s_modf[tid] = 0.25f    + atan2f(x7, x6) * (INV_PI * 4.75f);
    s_bb[tid]   = 0.1f     + atan2f(x9, x8) * (INV_PI * 9.9f);
  }
  if (tid < 32) __builtin_amdgcn_s_wait_tensorcnt(0);
  __syncthreads();

  // ---- 3) pass 1: per-thread chunk sums of the three phase increments -----
  const int base = tid * SPT;
  float sum1 = 0.f, sum2 = 0.f, sum3 = 0.f;   // f0+nz*std, mod_factor, f0
  for (int j = 0; j < SPT; ++j) {
    int i = base + j;
    float coords = (i + 0.5f) * (1.0f/256.0f) - 0.5f;
    coords = fminf(fmaxf(coords, 0.0f), 127.0f);
    int lo = (int)coords;
    int hi = min(lo + 1, N_FRAMES - 1);
    float w = coords - (float)lo;
    float f0i  = s_f0[lo]   + (s_f0[hi]   - s_f0[lo])   * w;
    float stdi = s_std[lo]  + (s_std[hi]  - s_std[lo])  * w;
    float mfi  = s_modf[lo] + (s_modf[hi] - s_modf[lo]) * w;
    float nz = s_noise[i];
    sum1 += f0i + nz * stdi;
    sum2 += mfi;
    sum3 += f0i;
  }

  // ---- 4) intra-wave inclusive scan (f32, shuffle) ------------------------
  float inc1 = sum1, inc2 = sum2, inc3 = sum3;
  #pragma unroll
  for (int d = 1; d < 32; d <<= 1) {
    float o1 = __shfl_up(inc1, d, 32);
    float o2 = __shfl_up(inc2, d, 32);
    float o3 = __shfl_up(inc3, d, 32);
    if (lane >= d) { inc1 += o1; inc2 += o2; inc3 += o3; }
  }
  if (lane == 31) {
    s_wtot[0][wave] = inc1; s_wtot[1][wave] = inc2; s_wtot[2][wave] = inc3;
  }
  __syncthreads();

  // ---- 5) inter-wave scan via WMMA (wave 0): D = L16 x V, f32 exact -------
  // V's 6 used columns = {q0_lo,q0_hi,q1_lo,q1_hi,q2_lo,q2_hi} of wave totals.
  if (tid < 32) {
    const int m_n   = lane & 15;     // row of A / column of B and D
    const int halfq = lane >> 4;     // selects K pair {0,1} vs {2,3}
    const int kb    = halfq * 2;
    const bool valid = (m_n < 6);
    const int q = valid ? (m_n >> 1) : 0;
    const int h = m_n & 1;
    v8f acc = {0.f,0.f,0.f,0.f,0.f,0.f,0.f,0.f};
    #pragma unroll
    for (int cc = 0; cc < 4; ++cc) {
      v2f a, bv;
      // A = columns [4cc,4cc+4) of lower-triangular ones: A[m][k] = (m >= k)
      a.x = (m_n >= 4*cc + kb    ) ? 1.0f : 0.0f;
      a.y = (m_n >= 4*cc + kb + 1) ? 1.0f : 0.0f;
      int w0 = h*16 + 4*cc + kb;     // wave index into the totals
      bv.x = valid ? s_wtot[q][w0    ] : 0.0f;
      bv.y = valid ? s_wtot[q][w0 + 1] : 0.0f;
      acc = __builtin_amdgcn_wmma_f32_16x16x4_f32(
          false, a, false, bv, (short)0, acc, false, false);
    }
    if (valid) {
      #pragma unroll
      for (int r = 0; r < 8; ++r)    // lane holds rows m = halfq*8 + r of col m_n
        s_wscan[q][h*16 + halfq*8 + r] = acc[r];
    }
  }
  __syncthreads();

  // ---- 6) combine: thread-exclusive prefix bases --------------------------
  float e1, e2, e3;
  {
    float off1 = 0.f, off2 = 0.f, off3 = 0.f;
    if (wave > 0) {
      int pw = wave - 1;
      float c1 = (pw >= 16) ? s_wscan[0][15] : 0.f;
      float c2 = (pw >= 16) ? s_wscan[1][15] : 0.f;
      float c3 = (pw >= 16) ? s_wscan[2][15] : 0.f;
      off1 = s_wscan[0][pw] + c1;
      off2 = s_wscan[1][pw] + c2;
      off3 = s_wscan[2][pw] + c3;
    }
    e1 = off1 + (inc1 - sum1);   // + exclusive-within-wave
    e2 = off2 + (inc2 - sum2);
    e3 = off3 + (inc3 - sum3);
  }

  // ---- 7) pass 2: running phases -> synth -> atomic channel mean ----------
  float r1 = e1, r2 = e2, r3 = e3;
  for (int j = 0; j < SPT; ++j) {
    int i = base + j;
    float coords = (i + 0.5f) * (1.0f/256.0f) - 0.5f;
    coords = fminf(fmaxf(coords, 0.0f), 127.0f);
    int lo = (int)coords;
    int hi = min(lo + 1, N_FRAMES - 1);
    float w = coords - (float)lo;
    float f0i  = s_f0[lo]   + (s_f0[hi]   - s_f0[lo])   * w;
    float stdi = s_std[lo]  + (s_std[hi]  - s_std[lo])  * w;
    float mfi  = s_modf[lo] + (s_modf[hi] - s_modf[lo]) * w;
    float ampi = s_amp[lo]  + (s_amp[hi]  - s_amp[lo])  * w;
    float mixi = s_mix[lo]  + (s_mix[hi]  - s_mix[lo])  * w;
    float bbi  = s_bb[lo]   + (s_bb[hi]   - s_bb[lo])   * w;
    float nz = s_noise[i];
    r1 += f0i + nz * stdi;        // phase of noise oscillator
    r2 += mfi;                    // phase of modulator
    r3 += f0i;                    // phase of carrier
    float harm = ampi * mixi;
    float namp = ampi * (1.0f - mixi);
    float mixed = namp * sinf(r1) + harm * sinf(r3 + bbi * sinf(r2));
    atomicAdd(&out[(size_t)b * N_SAMPLES + i], mixed * 0.125f);
  }
}

__global__ void zero_out_kernel(float* __restrict__ out, int n) {
  int i = blockIdx.x * blockDim.x + threadIdx.x;
  if (i < n) out[i] = 0.0f;
}

extern "C" void kernel_launch(void* const* d_in, const int* in_sizes, int n_in,
                              void* d_out, int out_size, void* d_ws, size_t ws_size,
                              hipStream_t stream) {
  const float* x     = (const float*)d_in[0];   // (16,8,10,128)
  const float* noise = (const float*)d_in[1];   // (16,8,32768)
  float* out = (float*)d_out;                   // (16,1,32768)

  zero_out_kernel<<<(out_size + THREADS - 1) / THREADS, THREADS, 0, stream>>>(out, out_size);
  fm_synth_kernel<<<128, THREADS, N_SAMPLES * sizeof(float), stream>>>(x, noise, out);
}